// Expand_24386824307320
// MI455X (gfx1250) — compile-verified
//
#include <hip/hip_runtime.h>
#include <stdint.h>

// Problem constants (from the reference file)
#define BB 16
#define SS 4096
#define DD 256
#define RPW 4      // output rows per wave (one LDS staging slot each)

typedef __attribute__((ext_vector_type(4))) float v4f;   // native vector for nontemporal builtin

// -----------------------------------------------------------------------------
// Kernel 1: per-batch inclusive prefix sum of durations.
// One block per batch, 256 threads, each thread owns 16 consecutive elements.
// -----------------------------------------------------------------------------
__global__ void __launch_bounds__(256)
duration_scan_kernel(const int* __restrict__ dims, int* __restrict__ csum) {
    const int b   = blockIdx.x;
    const int tid = threadIdx.x;                  // 0..255
    const int* __restrict__ din = dims + (size_t)b * SS;
    int* __restrict__ cout      = csum + (size_t)b * SS;

    int vals[16];
    int local = 0;
#pragma unroll
    for (int j = 0; j < 16; ++j) {
        local += din[tid * 16 + j];
        vals[j] = local;                           // inclusive within chunk
    }

    __shared__ int sums[256];
    sums[tid] = local;
    __syncthreads();
    // Hillis-Steele inclusive scan over the 256 per-thread totals
    for (int off = 1; off < 256; off <<= 1) {
        int v = (tid >= off) ? sums[tid - off] : 0;
        __syncthreads();
        sums[tid] += v;
        __syncthreads();
    }
    const int excl = sums[tid] - local;            // exclusive prefix of chunk
#pragma unroll
    for (int j = 0; j < 16; ++j) cout[tid * 16 + j] = excl + vals[j];
}

// -----------------------------------------------------------------------------
// Kernel 2: expand/gather. One wave32 per group of RPW consecutive output rows.
// Phase 1: per row, wave-uniform binary search of the cumsum (seeded by the
//          previous row's index -> near-free for duplicated rows), then issue
//          2 async b128 loads HBM -> private LDS slot (ASYNCcnt, no VGPR data).
// Single s_wait_asynccnt 0x0 for the whole group.
// Phase 2: per row, 2 async b128 stores LDS -> HBM with TH_STORE_NT (output is
//          write-once; keep x resident in L2 for the repeated gather hits);
//          padding rows write zeros with non-temporal VGPR stores.
// -----------------------------------------------------------------------------
__global__ void __launch_bounds__(256)
expand_gather_kernel(const float* __restrict__ x, const int* __restrict__ csum,
                     float* __restrict__ out, int T) {
    __shared__ float stage[8][RPW][DD];            // 4 KB staging per wave

    const int lane = threadIdx.x & 31;
    const int wave = threadIdx.x >> 5;
    const long long nrows = (long long)BB * (long long)T;
    const long long row0  = ((long long)blockIdx.x * 8 + wave) * RPW;
    if (row0 >= nrows) return;                     // wave-uniform exit

    int  valid[RPW];
    float* orows[RPW];

    int prev_b = -1;
    int total  = 0;
    int idx    = 0;

    // ---- Phase 1: search + issue async loads -------------------------------
#pragma unroll
    for (int j = 0; j < RPW; ++j) {
        const long long row = row0 + j;
        valid[j] = 0;
        orows[j] = nullptr;
        if (row >= nrows) continue;                // uniform across the wave

        const int b = (int)(row / T);
        const int t = (int)(row - (long long)b * T);
        if (b != prev_b) {                         // crossed a batch boundary
            total  = csum[(size_t)b * SS + (SS - 1)];
            prev_b = b;
            idx    = 0;                            // reset monotone seed
        }
        orows[j] = out + ((size_t)b * (size_t)T + (size_t)t) * DD;

        if (t < total) {
            // first i with csum[i] > t (searchsorted 'right'); idx monotone in t
            const int* __restrict__ c = csum + (size_t)b * SS;
            int lo = idx, hi = SS - 1;
            while (lo < hi) {
                int mid = (lo + hi) >> 1;
                if (c[mid] > t) hi = mid; else lo = mid + 1;
            }
            idx = lo;
            valid[j] = 1;

            const float* __restrict__ srow =
                x + ((size_t)b * SS + (size_t)lo) * DD;
            unsigned lds0 = (unsigned)(uintptr_t)&stage[wave][j][lane * 4];
            unsigned lds1 = lds0 + 512u;
            const float* g0 = srow + lane * 4;
            const float* g1 = g0 + 128;
            asm volatile("global_load_async_to_lds_b128 %0, %1, off"
                         :: "v"(lds0), "v"(g0) : "memory");
            asm volatile("global_load_async_to_lds_b128 %0, %1, off"
                         :: "v"(lds1), "v"(g1) : "memory");
        }
    }

    // ---- one wait for all outstanding async loads --------------------------
    asm volatile("s_wait_asynccnt 0x0" ::: "memory");

    // ---- Phase 2: drain LDS -> HBM (NT) or write zero padding --------------
#pragma unroll
    for (int j = 0; j < RPW; ++j) {
        if (orows[j] == nullptr) continue;
        if (valid[j]) {
            unsigned lds0 = (unsigned)(uintptr_t)&stage[wave][j][lane * 4];
            unsigned lds1 = lds0 + 512u;
            float* o0 = orows[j] + lane * 4;
            float* o1 = o0 + 128;
            asm volatile("global_store_async_from_lds_b128 %0, %1, off th:TH_STORE_NT"
                         :: "v"(o0), "v"(lds0) : "memory");
            asm volatile("global_store_async_from_lds_b128 %0, %1, off th:TH_STORE_NT"
                         :: "v"(o1), "v"(lds1) : "memory");
        } else {
            const v4f z = {0.f, 0.f, 0.f, 0.f};
            v4f* o = (v4f*)orows[j];
            __builtin_nontemporal_store(z, &o[lane]);
            __builtin_nontemporal_store(z, &o[lane + 32]);
        }
    }
    // outstanding async stores drain at S_ENDPGM (implicit wait-idle)
}

// -----------------------------------------------------------------------------
// Launch
// -----------------------------------------------------------------------------
extern "C" void kernel_launch(void* const* d_in, const int* in_sizes, int n_in,
                              void* d_out, int out_size, void* d_ws, size_t ws_size,
                              hipStream_t stream) {
    const float* x    = (const float*)d_in[0];     // [B,S,D] float32
    const int*   dims = (const int*)d_in[1];       // [B,S,1] int32
    float*       out  = (float*)d_out;             // [B,T,D] float32
    int*         csum = (int*)d_ws;                // [B,S] int32 scratch (256 KB)

    const int T = out_size / (BB * DD);            // padded output length

    duration_scan_kernel<<<BB, 256, 0, stream>>>(dims, csum);

    const long long nrows  = (long long)BB * (long long)T;
    const long long rows_per_block = 8LL * RPW;    // 8 waves x RPW rows
    const int blocks = (int)((nrows + rows_per_block - 1) / rows_per_block);
    expand_gather_kernel<<<blocks, 256, 0, stream>>>(x, csum, out, T);

    (void)in_sizes; (void)n_in; (void)ws_size;
}